// HashEncoder_8315056685330
// MI455X (gfx1250) — compile-verified
//
#include <hip/hip_runtime.h>
#include <stdint.h>

typedef float f32x2 __attribute__((ext_vector_type(2)));

namespace {
constexpr uint32_t P1 = 2654435761u;
constexpr uint32_t P2 = 805459861u;
constexpr uint32_t HMASK = (1u << 19) - 1u;   // HASHMAP - 1
// Row offsets of each level: dense 16^3, 32^3, 64^3, then 13 x 2^19 hashed.
constexpr int OFFS[16] = {
    0,       4096,    36864,   299008,  823296,  1347584, 1871872, 2396160,
    2920448, 3444736, 3969024, 4493312, 5017600, 5541888, 6066176, 6590464};
constexpr int LDS_STRIDE = 34;  // 32 floats/point + 2 pad (even -> b64-aligned, conflict-free)
}  // namespace

__global__ __launch_bounds__(256, 1) void hashenc16_kernel(
    const float* __restrict__ xg, const f32x2* __restrict__ emb,
    f32x2* __restrict__ out, long long npts) {
  __shared__ float smem[256 * LDS_STRIDE];

  const int t = threadIdx.x;
  long long b = (long long)blockIdx.x * 256 + t;
  const bool valid = b < npts;
  if (!valid) b = npts - 1;  // clamp for safe loads; store phase re-checks range

  // Streamed-once coordinates: non-temporal so they don't pollute L2
  // (L2 must stay resident with the 57MB embedding table).
  const float px = __builtin_nontemporal_load(xg + b * 3 + 0);
  const float py = __builtin_nontemporal_load(xg + b * 3 + 1);
  const float pz = __builtin_nontemporal_load(xg + b * 3 + 2);

  // Prefetch the next block's coordinates (gfx1250 global_prefetch_b8).
  long long pfi = b * 3 + 3 * 256;
  long long pfmax = npts * 3 - 1;
  __builtin_prefetch(xg + (pfi < pfmax ? pfi : pfmax), 0, 0);

#pragma unroll
  for (int l = 0; l < 16; ++l) {
    const int res = 16 << l;          // ceil(16 * 2^l) == 16<<l exactly
    const int hi = res - 1;
    const float rf = (float)hi;
    const int off = OFFS[l];
    const bool dense = (l < 3);       // 16^3, 32^3, 64^3 <= 2^19

    const float fx = px * rf, fy = py * rf, fz = pz * rf;
    const float gx = floorf(fx), gy = floorf(fy), gz = floorf(fz);
    const float w1x = fx - gx, w1y = fy - gy, w1z = fz - gz;
    const float w0x = 1.0f - w1x, w0y = 1.0f - w1y, w0z = 1.0f - w1z;

    const int ix = (int)gx, iy = (int)gy, iz = (int)gz;
    const int x0 = ix < 0 ? 0 : (ix > hi ? hi : ix);
    const int x1 = (ix + 1) < 0 ? 0 : ((ix + 1) > hi ? hi : (ix + 1));
    const int y0 = iy < 0 ? 0 : (iy > hi ? hi : iy);
    const int y1 = (iy + 1) < 0 ? 0 : ((iy + 1) > hi ? hi : (iy + 1));
    const int z0 = iz < 0 ? 0 : (iz > hi ? hi : iz);
    const int z1 = (iz + 1) < 0 ? 0 : ((iz + 1) > hi ? hi : (iz + 1));

    // Per-dimension index contributions hoisted out of the 8-corner loop.
    uint32_t ax0, ax1, ay0, ay1, az0, az1;
    if (dense) {
      ax0 = (uint32_t)x0;               ax1 = (uint32_t)x1;
      ay0 = (uint32_t)(y0 * res);       ay1 = (uint32_t)(y1 * res);
      az0 = (uint32_t)(z0 * res * res); az1 = (uint32_t)(z1 * res * res);
    } else {
      ax0 = (uint32_t)x0;               ax1 = (uint32_t)x1;  // prime 1
      ay0 = (uint32_t)y0 * P1;          ay1 = (uint32_t)y1 * P1;
      az0 = (uint32_t)z0 * P2;          az1 = (uint32_t)z1 * P2;
    }

    uint32_t idx[8];
#pragma unroll
    for (int k = 0; k < 8; ++k) {
      // itertools.product order: bit2 -> dim0(x), bit1 -> dim1(y), bit0 -> dim2(z)
      const uint32_t cx = (k & 4) ? ax1 : ax0;
      const uint32_t cy = (k & 2) ? ay1 : ay0;
      const uint32_t cz = (k & 1) ? az1 : az0;
      idx[k] = dense ? (cx + cy + cz) : ((cx ^ cy ^ cz) & HMASK);
    }

    // Issue all 8 gathers (L2-resident table, default RT hint) before use.
    f32x2 g[8];
#pragma unroll
    for (int k = 0; k < 8; ++k) g[k] = emb[off + (int)idx[k]];

    float s0 = 0.0f, s1 = 0.0f;
#pragma unroll
    for (int k = 0; k < 8; ++k) {
      const float w = ((k & 4) ? w1x : w0x) * ((k & 2) ? w1y : w0y) *
                      ((k & 1) ? w1z : w0z);
      s0 += w * g[k].x;
      s1 += w * g[k].y;
    }
    smem[t * LDS_STRIDE + 2 * l + 0] = s0;
    smem[t * LDS_STRIDE + 2 * l + 1] = s1;
  }

  __syncthreads();

  // Coalesced non-temporal writeout: transpose [256 pts][32 f32] through LDS
  // so each wave stores 256 contiguous bytes (full cachelines), bypassing L2
  // retention for the 256MB write-once output.
  const long long base2 = (long long)blockIdx.x * 4096;  // float2 index
  const long long lim2 = npts * 16;
#pragma unroll
  for (int i = 0; i < 16; ++i) {
    const int flat2 = i * 256 + t;
    if (base2 + flat2 < lim2) {
      const int p = flat2 >> 4;          // point within block
      const int c = (flat2 & 15) << 1;   // float offset within point's 32
      f32x2 v;
      v.x = smem[p * LDS_STRIDE + c + 0];
      v.y = smem[p * LDS_STRIDE + c + 1];
      __builtin_nontemporal_store(v, out + base2 + flat2);
    }
  }
}

extern "C" void kernel_launch(void* const* d_in, const int* in_sizes, int n_in,
                              void* d_out, int out_size, void* d_ws,
                              size_t ws_size, hipStream_t stream) {
  const float* x = (const float*)d_in[0];        // [B, 3] f32
  const f32x2* emb = (const f32x2*)d_in[1];      // [TOTAL_ROWS, 2] f32
  f32x2* out = (f32x2*)d_out;                    // [B, 32] f32
  const long long npts = (long long)in_sizes[0] / 3;
  const int blocks = (int)((npts + 255) / 256);
  hashenc16_kernel<<<blocks, 256, 0, stream>>>(x, emb, out, npts);
}